// Planner_39195871543742
// MI455X (gfx1250) — compile-verified
//
#include <hip/hip_runtime.h>
#include <hip/hip_bf16.h>
#include <math.h>

// ---------------------------------------------------------------------------
// Batched augmented-Lagrangian Newton solver (B=4096 independent samples).
//
// MI455X reasoning (compile-only):
//  * All data (x: 114KB, W1: 7KB, W2: 8KB) is cache-resident -> latency bound,
//    not 23.3TB/s HBM bound. The only matrix work is the 7->256->8 MLP, so we
//    batch 16 samples into the WMMA M dimension and use the *full-f32* matrix
//    pipe V_WMMA_F32_16X16X4_F32 (no precision downgrade inside Newton).
//  * MLP Jacobian is reformulated as G = P^T g with P[c][o*7+j]=W2[c,o]*W1[j,c]
//    precomputed in LDS (57KB) -> one extra 16x56x256 WMMA matmul per group.
//  * tanh uses CDNA5's hardware V_TANH_F32 (TRANS op, co-executes with VALU)
//    when the builtin exists; else exp + v_rcp_f32.
//  * Contiguous weight staging (W1 rows, b1) uses the gfx1250 async
//    GLOBAL_LOAD_ASYNC_TO_LDS_B128 path + s_wait_asynccnt (ASYNCcnt).
//  * Quaternion chain rule (E matrix, R block of J_phi) is closed-form
//    forward-mode per lane; 6x6 Cholesky solve + Armijo line search per lane
//    (wave32 -> lane == sample, 32 samples/wave).
//  * kappa update deferred into next iteration's forward pass (identical
//    algebra); global batch `done` flag folded out (post-convergence only).
// ---------------------------------------------------------------------------

typedef __attribute__((ext_vector_type(2))) float v2f;
typedef __attribute__((ext_vector_type(8))) float v8f;
typedef int v4i_n __attribute__((vector_size(16)));  // matches builtin param type
typedef __attribute__((address_space(1))) v4i_n* gptr_v4i;
typedef __attribute__((address_space(3))) v4i_n* lptr_v4i;

#define HIDc   256
#define DHc    8
#define NUf    10.0f
#define MAXITc 10
#define LSNc   10
#define BLOCKc 128   // 4 wave32 -> 128 samples per block

// ---- dynamic-LDS layout (in floats) ----
constexpr int TSs    = 257;                 // t row stride (odd -> no 32-way bank conflicts)
constexpr int GSSs   = 57;                  // G row stride
constexpr int OFF_W1 = 0;                   // [8][256]   (row 7 zero pad)
constexpr int OFF_W2 = OFF_W1 + 8 * HIDc;   // [256][16]  (cols 8..15 zero)
constexpr int OFF_P  = OFF_W2 + HIDc * 16;  // [256][64]  (cols 56..63 zero)
constexpr int OFF_B1 = OFF_P + HIDc * 64;   // [256]
constexpr int OFF_B2 = OFF_B1 + HIDc;       // [16]
constexpr int OFF_WV = OFF_B2 + 16;         // per-wave regions
constexpr int WV_X   = 0;                   // [32][8]   staged x (padded)
constexpr int WV_T   = WV_X + 32 * 8;       // [32][TSs] staged tanh activations
constexpr int WV_G   = WV_T + 32 * TSs;     // [32][57]  staged G
constexpr int WV_H   = WV_G + 32 * GSSs;    // [32][8]   staged h
constexpr int WVF    = WV_H + 32 * DHc;     // per-wave float count (10560)
constexpr int NWAVES = BLOCKc / 32;
constexpr int LDS_FLOATS = OFF_WV + NWAVES * WVF;  // 65040 floats = 254.1 KB

#ifndef __has_builtin
#define __has_builtin(x) 0
#endif

// ---------------------------------------------------------------------------
// WMMA: D = A(16x4 f32) * B(4x16 f32) + C(16x16 f32)
// ---------------------------------------------------------------------------
static __device__ __forceinline__ v8f wmma4(v2f a, v2f b, v8f c) {
#if __has_builtin(__builtin_amdgcn_wmma_f32_16x16x4_f32)
  // 8 args: (neg_a, A, neg_b, B, c_mod, C, reuse_a, reuse_b)
  return __builtin_amdgcn_wmma_f32_16x16x4_f32(false, a, false, b, (short)0, c,
                                               false, false);
#else
  v8f d;
  asm volatile("v_wmma_f32_16x16x4_f32 %0, %1, %2, %3"
               : "=v"(d)
               : "v"(a), "v"(b), "v"(c));
  return d;
#endif
}

// tanh: prefer CDNA5 hardware V_TANH_F32 (single TRANS instruction).
static __device__ __forceinline__ float fast_tanh(float v) {
#if __has_builtin(__builtin_amdgcn_tanhf)
  return __builtin_amdgcn_tanhf(v);
#elif __has_builtin(__builtin_amdgcn_tanh_f32)
  return __builtin_amdgcn_tanh_f32(v);
#elif __has_builtin(__builtin_amdgcn_rcpf)
  // tanh(v) = 1 - 2/(exp(2v)+1): v_exp + v_rcp, saturates correctly at +-inf.
  float e = __expf(2.0f * v);
  return 1.0f - 2.0f * __builtin_amdgcn_rcpf(e + 1.0f);
#else
  float e = __expf(2.0f * v);
  return 1.0f - 2.0f / (e + 1.0f);
#endif
}

// ---------------------------------------------------------------------------
// Per-lane quaternion math (wxyz, matches the reference exactly incl. eps)
// ---------------------------------------------------------------------------
static __device__ __forceinline__ void qmul(const float* a, const float* b, float* o) {
  o[0] = a[0] * b[0] - a[1] * b[1] - a[2] * b[2] - a[3] * b[3];
  o[1] = a[0] * b[1] + a[1] * b[0] + a[2] * b[3] - a[3] * b[2];
  o[2] = a[0] * b[2] - a[1] * b[3] + a[2] * b[0] + a[3] * b[1];
  o[3] = a[0] * b[3] + a[1] * b[2] - a[2] * b[1] + a[3] * b[0];
}

static __device__ __forceinline__ void aa2q(const float* aa, float* q) {
  float n2  = aa[0] * aa[0] + aa[1] * aa[1] + aa[2] * aa[2] + 1e-12f;
  float ang = sqrtf(n2);
  float hf  = 0.5f * ang;
  float s   = (ang < 1e-6f) ? (0.5f - ang * ang * (1.0f / 48.0f)) : (sinf(hf) / ang);
  q[0] = cosf(hf);
  q[1] = aa[0] * s; q[2] = aa[1] * s; q[3] = aa[2] * s;
}

static __device__ __forceinline__ void apply_delta(const float* x, const float* d,
                                                   float a, float* o) {
  o[0] = x[0] + a * d[0]; o[1] = x[1] + a * d[1]; o[2] = x[2] + a * d[2];
  float aa[3] = {a * d[3], a * d[4], a * d[5]};
  float qd[4], q[4];
  aa2q(aa, qd);
  qmul(x + 3, qd, q);
  o[3] = q[0]; o[4] = q[1]; o[5] = q[2]; o[6] = q[3];
}

static __device__ __forceinline__ void q2aa(const float* q, float* r) {
  float n   = sqrtf(q[1] * q[1] + q[2] * q[2] + q[3] * q[3] + 1e-12f);
  float hf  = atan2f(n, q[0]);
  float ang = 2.0f * hf;
  float s   = (fabsf(ang) < 1e-6f) ? (0.5f - ang * ang * (1.0f / 48.0f))
                                   : (sinf(hf) / ang);
  float inv = 1.0f / s;
  r[0] = q[1] * inv; r[1] = q[2] * inv; r[2] = q[3] * inv;
}

// tangent of p (x) [0, 0.5*e_k] for k=0..2  (d/d aa_k of p (x) aa2q(aa) at aa=0)
static __device__ __forceinline__ void qtan(const float* p, float dq[3][4]) {
  float w = p[0], x = p[1], y = p[2], z = p[3];
  dq[0][0] = -0.5f * x; dq[0][1] =  0.5f * w; dq[0][2] =  0.5f * z; dq[0][3] = -0.5f * y;
  dq[1][0] = -0.5f * y; dq[1][1] = -0.5f * z; dq[1][2] =  0.5f * w; dq[1][3] =  0.5f * x;
  dq[2][0] = -0.5f * z; dq[2][1] =  0.5f * y; dq[2][2] = -0.5f * x; dq[2][3] =  0.5f * w;
}

// value + forward-mode JVP of quat_to_axis_angle for 3 tangents
static __device__ __forceinline__ void q2aa_jvp(const float* q, const float dq[3][4],
                                                float* r, float R[3][3]) {
  float w = q[0], vx = q[1], vy = q[2], vz = q[3];
  float n2  = vx * vx + vy * vy + vz * vz + 1e-12f;
  float n   = sqrtf(n2);
  float hf  = atan2f(n, w);
  float ang = 2.0f * hf;
  bool  sm  = fabsf(ang) < 1e-6f;
  float sh = sinf(hf), ch = cosf(hf);
  float s    = sm ? (0.5f - ang * ang * (1.0f / 48.0f)) : (sh / ang);
  float invs = 1.0f / s;
  r[0] = vx * invs; r[1] = vy * invs; r[2] = vz * invs;
  float den  = n2 + w * w;
  float dsdh = sm ? (-ang * (1.0f / 12.0f)) : ((hf * ch - sh) / (2.0f * hf * hf));
#pragma unroll
  for (int k = 0; k < 3; ++k) {
    float dw = dq[k][0], dvx = dq[k][1], dvy = dq[k][2], dvz = dq[k][3];
    float dn = (vx * dvx + vy * dvy + vz * dvz) / n;
    float dh = (w * dn - n * dw) / den;
    float ds = dsdh * dh;
    float t2 = ds * invs * invs;
    R[0][k] = dvx * invs - vx * t2;
    R[1][k] = dvy * invs - vy * t2;
    R[2][k] = dvz * invs - vz * t2;
  }
}

// ---------------------------------------------------------------------------
// Cooperative MLP evaluation for the wave's 32 samples (2 WMMA groups of 16).
// Produces h in hS (and, if withG, G = dh/dxp in GS). Uniform control flow.
// ---------------------------------------------------------------------------
static __device__ __forceinline__ void coop_eval(
    float* xS, float* tS, float* GS, float* hS,
    const float* W1s, const float* W2s, const float* Ps,
    const float* b1s, const float* b2s,
    const float xt[7], bool withG, int lane, int half, int lm, int ks) {
  // stage this lane's padded state
#pragma unroll
  for (int i = 0; i < 7; ++i) xS[lane * 8 + i] = xt[i];
  xS[lane * 8 + 7] = 0.0f;
  __syncthreads();

  // hidden layer: Z = X * W1 + b1 ; t = tanh(Z)  (A: 16 samples x 8, K split 2x4)
  for (int grp = 0; grp < 2; ++grp) {
    const float* xr = xS + (grp * 16 + lm) * 8;
    for (int nt = 0; nt < 16; ++nt) {
      v8f acc = {0.f, 0.f, 0.f, 0.f, 0.f, 0.f, 0.f, 0.f};
      int col = nt * 16 + lm;
#pragma unroll
      for (int kk = 0; kk < 2; ++kk) {
        v2f a, b;
        a.x = xr[kk * 4 + ks];
        a.y = xr[kk * 4 + ks + 1];
        int kb = kk * 4 + 2 * half;
        b.x = W1s[kb * HIDc + col];
        b.y = W1s[(kb + 1) * HIDc + col];
        acc = wmma4(a, b, acc);
      }
      float bb = b1s[col];
#pragma unroll
      for (int r = 0; r < 8; ++r)  // C/D layout: row = r + 8*half, col = lm
        tS[(grp * 16 + r + half * 8) * TSs + col] = fast_tanh(acc[r] + bb);
    }
  }
  __syncthreads();

  // output layer: h = t * W2 + b2   (K = 256 -> 64 WMMA K-steps)
  for (int grp = 0; grp < 2; ++grp) {
    const float* tr = tS + (grp * 16 + lm) * TSs;
    v8f acc = {0.f, 0.f, 0.f, 0.f, 0.f, 0.f, 0.f, 0.f};
    for (int kk = 0; kk < 64; ++kk) {
      v2f a, b;
      a.x = tr[kk * 4 + ks];
      a.y = tr[kk * 4 + ks + 1];
      int kb = kk * 4 + 2 * half;
      b.x = W2s[kb * 16 + lm];
      b.y = W2s[(kb + 1) * 16 + lm];
      acc = wmma4(a, b, acc);
    }
    if (lm < DHc) {
#pragma unroll
      for (int r = 0; r < 8; ++r)
        hS[(grp * 16 + r + half * 8) * DHc + lm] = acc[r] + b2s[lm];
    }
  }

  // G = diag(1-t^2) contraction with P:  G[m][o*7+j] = sum_c (1-t^2) P[c][o*7+j]
  if (withG) {
    for (int grp = 0; grp < 2; ++grp) {
      const float* tr = tS + (grp * 16 + lm) * TSs;
      for (int nt = 0; nt < 4; ++nt) {
        v8f acc = {0.f, 0.f, 0.f, 0.f, 0.f, 0.f, 0.f, 0.f};
        int col = nt * 16 + lm;
        for (int kk = 0; kk < 64; ++kk) {
          float t0 = tr[kk * 4 + ks];
          float t1 = tr[kk * 4 + ks + 1];
          v2f a, b;
          a.x = 1.0f - t0 * t0;
          a.y = 1.0f - t1 * t1;
          int kb = kk * 4 + 2 * half;
          b.x = Ps[kb * 64 + col];
          b.y = Ps[(kb + 1) * 64 + col];
          acc = wmma4(a, b, acc);
        }
        if (col < 56) {
#pragma unroll
          for (int r = 0; r < 8; ++r)
            GS[(grp * 16 + r + half * 8) * GSSs + col] = acc[r];
        }
      }
    }
  }
  __syncthreads();
}

// ---------------------------------------------------------------------------
__global__ void __launch_bounds__(BLOCKc)
planner_kernel(const float* __restrict__ X, const float* __restrict__ W1,
               const float* __restrict__ b1, const float* __restrict__ W2,
               const float* __restrict__ b2, float* __restrict__ out, int n) {
  extern __shared__ float smem[];
  float* W1s = smem + OFF_W1;
  float* W2s = smem + OFF_W2;
  float* Ps  = smem + OFF_P;
  float* b1s = smem + OFF_B1;
  float* b2s = smem + OFF_B2;

  const int tid  = threadIdx.x;
  const int lane = tid & 31;
  const int wv   = tid >> 5;
  float* wb = smem + OFF_WV + wv * WVF;
  float* xS = wb + WV_X;
  float* tS = wb + WV_T;
  float* GS = wb + WV_G;
  float* hS = wb + WV_H;
  const int half = lane >> 4;
  const int lm   = lane & 15;
  const int ks   = half * 2;

  // ---- stage weights ----
#if __has_builtin(__builtin_amdgcn_global_load_async_to_lds_b128) && \
    __has_builtin(__builtin_amdgcn_s_wait_asynccnt)
  // gfx1250 async DMA path: W1 rows 0..6 (1792 floats) and b1 (256 floats)
  // are contiguous copies -> GLOBAL_LOAD_ASYNC_TO_LDS_B128, tracked by ASYNCcnt.
  {
    for (int idx = tid; idx < 448 + 64; idx += BLOCKc) {
      const float* gsrc;
      float* ldst;
      if (idx < 448) { gsrc = W1 + idx * 4; ldst = W1s + idx * 4; }
      else           { gsrc = b1 + (idx - 448) * 4; ldst = b1s + (idx - 448) * 4; }
      __builtin_amdgcn_global_load_async_to_lds_b128(
          (gptr_v4i)gsrc, (lptr_v4i)ldst, 0, 0);
    }
    for (int idx = tid; idx < HIDc; idx += BLOCKc) W1s[7 * HIDc + idx] = 0.0f;
    __builtin_amdgcn_s_wait_asynccnt(0);
  }
#else
  for (int idx = tid; idx < 8 * HIDc; idx += BLOCKc) {
    int j = idx >> 8, c = idx & 255;
    W1s[idx] = (j < 7) ? W1[j * HIDc + c] : 0.0f;
  }
  for (int idx = tid; idx < HIDc; idx += BLOCKc) b1s[idx] = b1[idx];
#endif
  for (int idx = tid; idx < HIDc * 16; idx += BLOCKc) {
    int c = idx >> 4, o = idx & 15;
    W2s[idx] = (o < DHc) ? W2[c * DHc + o] : 0.0f;
  }
  for (int idx = tid; idx < HIDc * 64; idx += BLOCKc) {
    int c = idx >> 6, col = idx & 63;
    float v = 0.0f;
    if (col < 56) {
      int o = col / 7, j = col - 7 * o;
      v = W2[c * DHc + o] * W1[j * HIDc + c];
    }
    Ps[idx] = v;
  }
  for (int idx = tid; idx < 16; idx += BLOCKc) b2s[idx] = (idx < DHc) ? b2[idx] : 0.0f;
  __syncthreads();

  // ---- per-lane state (lane == sample) ----
  int sid  = blockIdx.x * BLOCKc + tid;
  int sidc = (sid < n) ? sid : (n - 1);
  float x0[7], x[7];
#pragma unroll
  for (int i = 0; i < 7; ++i) { x0[i] = X[sidc * 7 + i]; x[i] = x0[i]; }
  float kap[DHc];
#pragma unroll
  for (int o = 0; o < DHc; ++o) kap[o] = 0.0f;
  const float q0i[4] = {x0[3], -x0[4], -x0[5], -x0[6]};  // quat_invert(x0)

  for (int it = 0; it < MAXITc; ++it) {
    // full eval at x: h + G
    coop_eval(xS, tS, GS, hS, W1s, W2s, Ps, b1s, b2s, x, true, lane, half, lm, ks);
    float h[DHc];
#pragma unroll
    for (int o = 0; o < DHc; ++o) h[o] = hS[lane * DHc + o];
    if (it > 0) {  // deferred kappa update: kappa_k = kappa_{k-1} + 2*nu*h(x_k)
#pragma unroll
      for (int o = 0; o < DHc; ++o) kap[o] += 2.0f * NUf * h[o];
    }

    // phi value + R block of J_phi (forward-mode through q2aa)
    float Q[4];
    qmul(q0i, x + 3, Q);
    float dQ[3][4];
    qtan(Q, dQ);
    float pr[3], Rm[3][3];
    q2aa_jvp(Q, dQ, pr, Rm);
    float pp[3] = {x[0] - x0[0], x[1] - x0[1], x[2] - x0[2]};

    float f = pp[0] * pp[0] + pp[1] * pp[1] + pp[2] * pp[2] +
              pr[0] * pr[0] + pr[1] * pr[1] + pr[2] * pr[2];
#pragma unroll
    for (int o = 0; o < DHc; ++o) f += h[o] * kap[o] + NUf * h[o] * h[o];

    // g = 2*J_phi*phi (reference: no transpose) + J_h^T(2 nu h + kappa)
    // H = 2*J_phi^T J_phi + 2 nu J_h^T J_h
    float g[6], Hm[6][6];
#pragma unroll
    for (int i = 0; i < 6; ++i) {
#pragma unroll
      for (int j = 0; j < 6; ++j) Hm[i][j] = 0.0f;
    }
#pragma unroll
    for (int i = 0; i < 3; ++i) {
      g[i] = 2.0f * pp[i];
      Hm[i][i] = 2.0f;
    }
#pragma unroll
    for (int k = 0; k < 3; ++k)
      g[3 + k] = 2.0f * (Rm[k][0] * pr[0] + Rm[k][1] * pr[1] + Rm[k][2] * pr[2]);
#pragma unroll
    for (int a = 0; a < 3; ++a) {
#pragma unroll
      for (int b = 0; b < 3; ++b) {
        float s = 0.0f;
#pragma unroll
        for (int k = 0; k < 3; ++k) s += Rm[k][a] * Rm[k][b];
        Hm[3 + a][3 + b] = 2.0f * s;
      }
    }
    float dqx[3][4];  // E quat block: d(x.q (x) aa2q(aa))/d aa at 0
    qtan(x + 3, dqx);
#pragma unroll
    for (int o = 0; o < DHc; ++o) {
      float Go[7];
#pragma unroll
      for (int j = 0; j < 7; ++j) Go[j] = GS[lane * GSSs + o * 7 + j];
      float Jh[6];
      Jh[0] = Go[0]; Jh[1] = Go[1]; Jh[2] = Go[2];
#pragma unroll
      for (int k = 0; k < 3; ++k)
        Jh[3 + k] = Go[3] * dqx[k][0] + Go[4] * dqx[k][1] +
                    Go[5] * dqx[k][2] + Go[6] * dqx[k][3];
      float gv = 2.0f * NUf * h[o] + kap[o];
#pragma unroll
      for (int i = 0; i < 6; ++i) {
        g[i] += Jh[i] * gv;
#pragma unroll
        for (int j = 0; j < 6; ++j) Hm[i][j] += 2.0f * NUf * Jh[i] * Jh[j];
      }
    }

    // Newton step: H delta = -g  (H SPD -> Cholesky, fully unrolled)
#pragma unroll
    for (int kc = 0; kc < 6; ++kc) {
      float d = Hm[kc][kc];
#pragma unroll
      for (int j = 0; j < kc; ++j) d -= Hm[kc][j] * Hm[kc][j];
      d = sqrtf(fmaxf(d, 1e-20f));
      Hm[kc][kc] = d;
      float inv = 1.0f / d;
#pragma unroll
      for (int i = kc + 1; i < 6; ++i) {
        float s = Hm[i][kc];
#pragma unroll
        for (int j = 0; j < kc; ++j) s -= Hm[i][j] * Hm[kc][j];
        Hm[i][kc] = s * inv;
      }
    }
    float yv[6], del[6];
#pragma unroll
    for (int i = 0; i < 6; ++i) {
      float s = -g[i];
#pragma unroll
      for (int j = 0; j < i; ++j) s -= Hm[i][j] * yv[j];
      yv[i] = s / Hm[i][i];
    }
#pragma unroll
    for (int i = 5; i >= 0; --i) {
      float s = yv[i];
#pragma unroll
      for (int j = i + 1; j < 6; ++j) s -= Hm[j][i] * del[j];
      del[i] = s / Hm[i][i];
    }

    float gd = 0.0f;
#pragma unroll
    for (int i = 0; i < 6; ++i) gd += g[i] * del[i];

    // Armijo backtracking (per-lane alpha, uniform loop -> WMMA EXEC all-1s)
    float alpha = 0.8f;
    for (int ls = 0; ls < LSNc; ++ls) {
      float xt[7];
      apply_delta(x, del, alpha, xt);
      coop_eval(xS, tS, GS, hS, W1s, W2s, Ps, b1s, b2s, xt, false, lane, half, lm, ks);
      float Qt[4], rt[3];
      qmul(q0i, xt + 3, Qt);
      q2aa(Qt, rt);
      float p0 = xt[0] - x0[0], p1 = xt[1] - x0[1], p2 = xt[2] - x0[2];
      float ft = p0 * p0 + p1 * p1 + p2 * p2 + rt[0] * rt[0] + rt[1] * rt[1] + rt[2] * rt[2];
#pragma unroll
      for (int o = 0; o < DHc; ++o) {
        float hv = hS[lane * DHc + o];
        ft += hv * kap[o] + NUf * hv * hv;
      }
      if (ft > f + 0.5f * gd * alpha) alpha *= 0.5f;
    }

    float xn[7];
    apply_delta(x, del, alpha, xn);
#pragma unroll
    for (int i = 0; i < 7; ++i) x[i] = xn[i];
  }

  if (sid < n) {
#pragma unroll
    for (int i = 0; i < 7; ++i) out[sid * 7 + i] = x[i];
  }
}

// ---------------------------------------------------------------------------
extern "C" void kernel_launch(void* const* d_in, const int* in_sizes, int n_in,
                              void* d_out, int out_size, void* d_ws, size_t ws_size,
                              hipStream_t stream) {
  (void)n_in; (void)out_size; (void)d_ws; (void)ws_size;
  const float* X  = (const float*)d_in[0];
  const float* W1 = (const float*)d_in[1];
  const float* b1 = (const float*)d_in[2];
  const float* W2 = (const float*)d_in[3];
  const float* b2 = (const float*)d_in[4];
  float* out = (float*)d_out;
  int n = in_sizes[0] / 7;
  if (n <= 0) return;
  int nblk = (n + BLOCKc - 1) / BLOCKc;
  size_t smem = (size_t)LDS_FLOATS * sizeof(float);
  hipLaunchKernelGGL(planner_kernel, dim3(nblk), dim3(BLOCKc), smem, stream,
                     X, W1, b1, W2, b2, out, n);
}